// BasicBlock_21131239097098
// MI455X (gfx1250) — compile-verified
//
#include <hip/hip_runtime.h>
#include <hip/hip_bf16.h>

// ---------------------------------------------------------------------------
// AdderNet BasicBlock for MI455X (gfx1250, wave32, WMMA f32 16x16x4).
//   repack : w_shift2 [co][ci][3][3] -> [kidx][co][ci]        (GEMM-friendly A)
//   stage1 : conv1x1 GEMM, LDS-staged B, M=32 reg blocking    x  -> t1
//   stage2 : adder1x1 + BN1 + ReLU (VALU, LDS weights)        t1 -> t2
//   stage3 : conv3x3 pad1 = 9 GEMM passes, LDS-staged B       t2 -> t3
//   stage4 : adder3x3 pad1 + BN2 + ReLU + residual + ReLU     t3 -> out
// Shapes fixed by reference: N=4, C=128, H=W=28 (784 = 49*16 px tiles exact).
// ---------------------------------------------------------------------------

typedef __attribute__((ext_vector_type(2))) float v2f;
typedef __attribute__((ext_vector_type(8))) float v8f;

#define NPIX 784              // 28*28 == 49*16
#define CCH  128
#define CHW  (CCH * NPIX)

// ---------------------------------------------------------------------------
// Repack 3x3 weights: W2r[kidx][co][ci] = W2[co][ci][kh][kw], kidx = kh*3+kw.
// Makes the A-fragment a contiguous 64-bit load in the GEMM K dimension.
// ---------------------------------------------------------------------------
__global__ __launch_bounds__(256) void repack_w3x3(const float* __restrict__ W2,
                                                   float* __restrict__ W2r) {
    int idx = blockIdx.x * 256 + threadIdx.x;           // over 128*128*9
    if (idx >= CCH * CCH * 9) return;
    int kidx = idx % 9;
    int ci   = (idx / 9) % CCH;
    int co   = idx / (9 * CCH);
    W2r[(size_t)kidx * CCH * CCH + (size_t)co * CCH + ci] = W2[idx];
}

// ---------------------------------------------------------------------------
// Stage 1: 1x1 conv as GEMM.  Y[co,p] = sum_ci W[co,ci] * X[ci,p]  (per batch)
// Block: 128 threads = 4 waves. Block owns one 16-pixel tile; the 128x16
// activation tile is staged once in LDS and shared. Each wave computes a
// 32(co) x 16(px) output tile: 2 WMMA accumulators sharing one B fragment.
// ---------------------------------------------------------------------------
__global__ __launch_bounds__(128) void conv1x1_wmma(const float* __restrict__ X,
                                                    const float* __restrict__ W,
                                                    float* __restrict__ Y) {
    const int p0   = blockIdx.x * 16;       // pixel tile (GEMM N)
    const int n    = blockIdx.y;            // batch
    const int wave = threadIdx.x >> 5;
    const int lane = threadIdx.x & 31;
    const int half = lane >> 4;             // selects K pair {0,1} vs {2,3}
    const int ln   = lane & 15;             // M row (A) / N col (B)
    const int co0  = wave * 32;             // this wave's 32-channel strip

    const float* Xn = X + (size_t)n * CHW;

    // ---- stage B tile: sh[ci][px], 128x16 f32 = 8KB, coalesced 16-wide ----
    __shared__ float sh[CCH][16];
    {
        const int px  = threadIdx.x & 15;
        const int cib = threadIdx.x >> 4;   // 8 ci rows per pass
        for (int ci = cib; ci < CCH; ci += 8)
            sh[ci][px] = Xn[(size_t)ci * NPIX + p0 + px];
    }
    __syncthreads();

    const float* Wr0 = W + (size_t)(co0 + ln)      * CCH;   // A rows, K-contig
    const float* Wr1 = W + (size_t)(co0 + 16 + ln) * CCH;
    __builtin_prefetch(Wr0, 0, 1);          // global_prefetch_b8

    v8f c0 = {}, c1 = {};
    for (int k = 0; k < CCH; k += 4) {
        const int k2 = k + 2 * half;
        v2f a0 = *(const v2f*)(Wr0 + k2);
        v2f a1 = *(const v2f*)(Wr1 + k2);
        v2f b;
        b.x = sh[k2][ln];
        b.y = sh[k2 + 1][ln];
        c0 = __builtin_amdgcn_wmma_f32_16x16x4_f32(false, a0, false, b,
                                                   (short)0, c0, false, false);
        c1 = __builtin_amdgcn_wmma_f32_16x16x4_f32(false, a1, false, b,
                                                   (short)0, c1, false, false);
    }

    float* Yn = Y + (size_t)n * CHW;
#pragma unroll
    for (int r = 0; r < 8; ++r) {           // D: VGPR r = rows r / r+8
        Yn[(size_t)(co0 + r + 8 * half)      * NPIX + p0 + ln] = c0[r];
        Yn[(size_t)(co0 + 16 + r + 8 * half) * NPIX + p0 + ln] = c1[r];
    }
}

// ---------------------------------------------------------------------------
// Stage 3: 3x3 conv pad=1 as 9 shifted GEMM passes over repacked weights.
// Halo masking (zero padding) is applied once per block while staging the
// shifted activation tile into LDS; the inner loop is pure ds_load + WMMA.
// ---------------------------------------------------------------------------
__global__ __launch_bounds__(128) void conv3x3_wmma(const float* __restrict__ X,
                                                    const float* __restrict__ W2r,
                                                    float* __restrict__ Y) {
    const int p0   = blockIdx.x * 16;
    const int n    = blockIdx.y;
    const int wave = threadIdx.x >> 5;
    const int lane = threadIdx.x & 31;
    const int half = lane >> 4;
    const int ln   = lane & 15;
    const int co0  = wave * 32;

    const float* Xn = X + (size_t)n * CHW;

    // staging-thread pixel geometry (computed once)
    const int px  = threadIdx.x & 15;
    const int cib = threadIdx.x >> 4;
    const int sp  = p0 + px;
    const int sho = sp / 28;
    const int swo = sp % 28;

    __shared__ float sh[CCH][16];

    v8f c0 = {}, c1 = {};
#pragma unroll
    for (int kidx = 0; kidx < 9; ++kidx) {
        const int dh = kidx / 3 - 1;
        const int dw = kidx % 3 - 1;

        // ---- restage shifted B tile with zero padding at the halo ----
        __syncthreads();                      // previous pass done reading
        {
            const int  ih = sho + dh;
            const int  iw = swo + dw;
            const bool ok = ((unsigned)ih < 28u) && ((unsigned)iw < 28u);
            const int  q  = ih * 28 + iw;
            for (int ci = cib; ci < CCH; ci += 8)
                sh[ci][px] = ok ? Xn[(size_t)ci * NPIX + q] : 0.f;
        }
        __syncthreads();

        // ---- GEMM pass: K = 128 over repacked weights [kidx][co][ci] ----
        const float* Wk  = W2r + (size_t)kidx * CCH * CCH;
        const float* Wr0 = Wk + (size_t)(co0 + ln)      * CCH;
        const float* Wr1 = Wk + (size_t)(co0 + 16 + ln) * CCH;
        for (int k = 0; k < CCH; k += 4) {
            const int k2 = k + 2 * half;
            v2f a0 = *(const v2f*)(Wr0 + k2);
            v2f a1 = *(const v2f*)(Wr1 + k2);
            v2f b;
            b.x = sh[k2][ln];
            b.y = sh[k2 + 1][ln];
            c0 = __builtin_amdgcn_wmma_f32_16x16x4_f32(false, a0, false, b,
                                                       (short)0, c0, false, false);
            c1 = __builtin_amdgcn_wmma_f32_16x16x4_f32(false, a1, false, b,
                                                       (short)0, c1, false, false);
        }
    }

    float* Yn = Y + (size_t)n * CHW;
#pragma unroll
    for (int r = 0; r < 8; ++r) {
        Yn[(size_t)(co0 + r + 8 * half)      * NPIX + p0 + ln] = c0[r];
        Yn[(size_t)(co0 + 16 + r + 8 * half) * NPIX + p0 + ln] = c1[r];
    }
}

// ---------------------------------------------------------------------------
// Stage 2: 1x1 adder (L1 distance) + BN1 + ReLU.
//   t2[n,co,p] = relu( (-sum_ci |t1[n,ci,p] - w[co,ci]|) * inv + beta )
// ---------------------------------------------------------------------------
__global__ __launch_bounds__(256) void adder1x1_bn_relu(
        const float* __restrict__ T1, const float* __restrict__ WA,
        const float* __restrict__ g,  const float* __restrict__ b,
        const float* __restrict__ m,  const float* __restrict__ v,
        float* __restrict__ T2) {
    const int co = blockIdx.x;
    const int n  = blockIdx.y;

    __shared__ float w[CCH];
    if (threadIdx.x < CCH) w[threadIdx.x] = WA[(size_t)co * CCH + threadIdx.x];
    __syncthreads();

    const float inv  = g[co] * rsqrtf(v[co] + 1e-5f);
    const float beta = b[co] - m[co] * inv;

    const float* Tn = T1 + (size_t)n * CHW;
    float*       On = T2 + ((size_t)n * CCH + co) * NPIX;

    for (int p = threadIdx.x; p < NPIX; p += 256) {
        float acc = 0.f;
#pragma unroll 8
        for (int ci = 0; ci < CCH; ++ci)
            acc += fabsf(Tn[(size_t)ci * NPIX + p] - w[ci]);
        On[p] = fmaxf(-acc * inv + beta, 0.f);
    }
}

// ---------------------------------------------------------------------------
// Stage 4: 3x3 adder pad=1 + BN2 + ReLU + residual add + ReLU.
// AdderNet padding: padded zeros contribute |0 - w| (taps are NOT skipped).
// ---------------------------------------------------------------------------
__global__ __launch_bounds__(256) void adder3x3_bn_relu_res(
        const float* __restrict__ T3, const float* __restrict__ WA,
        const float* __restrict__ g,  const float* __restrict__ b,
        const float* __restrict__ m,  const float* __restrict__ v,
        const float* __restrict__ XID, float* __restrict__ OUT) {
    const int co = blockIdx.x;
    const int n  = blockIdx.y;

    __shared__ float w[CCH * 9];
    for (int i = threadIdx.x; i < CCH * 9; i += 256)
        w[i] = WA[(size_t)co * (CCH * 9) + i];
    __syncthreads();

    const float inv  = g[co] * rsqrtf(v[co] + 1e-5f);
    const float beta = b[co] - m[co] * inv;

    const float* Tn  = T3  + (size_t)n * CHW;
    const float* idn = XID + ((size_t)n * CCH + co) * NPIX;
    float*       On  = OUT + ((size_t)n * CCH + co) * NPIX;

    for (int p = threadIdx.x; p < NPIX; p += 256) {
        const int ho = p / 28;
        const int wo = p % 28;
        float acc = 0.f;
        for (int ci = 0; ci < CCH; ++ci) {
            const float* xr = Tn + (size_t)ci * NPIX;
            const float* wr = w + ci * 9;
#pragma unroll
            for (int kh = 0; kh < 3; ++kh) {
                const int  ih = ho + kh - 1;
                const bool vh = (unsigned)ih < 28u;
#pragma unroll
                for (int kw = 0; kw < 3; ++kw) {
                    const int   iw = wo + kw - 1;
                    const bool  ok = vh && ((unsigned)iw < 28u);
                    const float xv = ok ? xr[ih * 28 + iw] : 0.f;
                    acc += fabsf(xv - wr[kh * 3 + kw]);
                }
            }
        }
        const float o = fmaxf(-acc * inv + beta, 0.f);
        On[p] = fmaxf(o + idn[p], 0.f);
    }
}

// ---------------------------------------------------------------------------
extern "C" void kernel_launch(void* const* d_in, const int* in_sizes, int n_in,
                              void* d_out, int out_size, void* d_ws, size_t ws_size,
                              hipStream_t stream) {
    const float* x        = (const float*)d_in[0];
    const float* w_shift1 = (const float*)d_in[1];
    const float* w_add1   = (const float*)d_in[2];
    const float* bn1_g    = (const float*)d_in[3];
    const float* bn1_b    = (const float*)d_in[4];
    const float* bn1_m    = (const float*)d_in[5];
    const float* bn1_v    = (const float*)d_in[6];
    const float* w_shift2 = (const float*)d_in[7];
    const float* w_add2   = (const float*)d_in[8];
    const float* bn2_g    = (const float*)d_in[9];
    const float* bn2_b    = (const float*)d_in[10];
    const float* bn2_m    = (const float*)d_in[11];
    const float* bn2_v    = (const float*)d_in[12];

    const size_t elems = (size_t)4 * CHW;       // 401408 floats per tensor
    float* t1  = (float*)d_ws;
    float* t2  = t1 + elems;
    float* t3  = t2 + elems;
    float* w2r = t3 + elems;                    // 128*128*9 = 147456 floats
    float* out = (float*)d_out;

    dim3 gemm_grid(NPIX / 16, 4);               // 49 pixel tiles x 4 batches
    dim3 ew_grid(CCH, 4);

    repack_w3x3<<<(CCH * CCH * 9 + 255) / 256, 256, 0, stream>>>(w_shift2, w2r);
    conv1x1_wmma<<<gemm_grid, 128, 0, stream>>>(x, w_shift1, t1);
    adder1x1_bn_relu<<<ew_grid, 256, 0, stream>>>(t1, w_add1, bn1_g, bn1_b,
                                                  bn1_m, bn1_v, t2);
    conv3x3_wmma<<<gemm_grid, 128, 0, stream>>>(t2, w2r, t3);
    adder3x3_bn_relu_res<<<ew_grid, 256, 0, stream>>>(t3, w_add2, bn2_g, bn2_b,
                                                      bn2_m, bn2_v, x, out);
}